// Attention_head_43825846288847
// MI455X (gfx1250) — compile-verified
//
#include <hip/hip_runtime.h>
#include <hip/hip_bf16.h>
#include <math.h>

// MI455X / gfx1250, wave32. Matrix ops via V_WMMA_F32_16X16X4_F32 (fp32-exact).

typedef __attribute__((ext_vector_type(2))) float v2f;
typedef __attribute__((ext_vector_type(8))) float v8f;

#define EMB 512
#define HD  64
#define SEQ 2048
#define NB  8
#define KC  32            // K-chunk for projection GEMM
#define XS_STRIDE 36      // 16-row x tile, padded (36*4B: bank-conflict-free, 16B aligned)
#define WT_STRIDE 34      // W^T cols-major stride (34: conflict-free, rows 8B aligned)
#define PB_STRIDE 18      // p-tile LDS stride (18 floats: conflict-free, 8B aligned)

__device__ __forceinline__ v8f wmma4(v2f a, v2f b, v8f c) {
  // D = A(16x4,f32) * B(4x16,f32) + C(16x16,f32)
  return __builtin_amdgcn_wmma_f32_16x16x4_f32(false, a, false, b, (short)0, c,
                                               false, false);
}

// Stage a KC x 64 chunk of a weight matrix into LDS *transposed* (WT[col][krow])
// so a B fragment (two consecutive K rows, same column) is one aligned b64 load.
__device__ __forceinline__ void stage_wT(const float* __restrict__ W, float* dst,
                                         int kc, int tid) {
#pragma unroll
  for (int i = 0; i < 4; ++i) {
    int vi = i * 128 + tid;      // 0..511 float4 slots (32 rows * 16 float4)
    int r  = vi >> 4;            // K row within chunk
    int c4 = vi & 15;
    const float4 val = *(const float4*)(W + (size_t)(kc + r) * HD + c4 * 4);
    dst[(c4 * 4 + 0) * WT_STRIDE + r] = val.x;
    dst[(c4 * 4 + 1) * WT_STRIDE + r] = val.y;
    dst[(c4 * 4 + 2) * WT_STRIDE + r] = val.z;
    dst[(c4 * 4 + 3) * WT_STRIDE + r] = val.w;
  }
}

// ---------------------------------------------------------------------------
// Kernel 1: q/k = x*Wq, x*Wk (row-major out); v^T = (x*Wv)^T ([B][64][2048]).
// One wave per 16-row tile of x (1024 tiles, 4 waves/block).
// ---------------------------------------------------------------------------
__global__ __launch_bounds__(128) void qkv_proj_kernel(
    const float* __restrict__ x,
    const float* __restrict__ Wq, const float* __restrict__ Wk,
    const float* __restrict__ Wv,
    float* __restrict__ q, float* __restrict__ k, float* __restrict__ vT) {
  __shared__ float xs[4][16 * XS_STRIDE];
  __shared__ float wt[3][HD * WT_STRIDE];

  const int tid  = threadIdx.x;
  const int wave = tid >> 5;
  const int lane = tid & 31;
  const int m    = lane & 15;   // A row / B col / D col
  const int hi   = lane >> 4;   // half-wave select
  const int row0 = (blockIdx.x * 4 + wave) * 16;   // global row in [0,16384)

  v8f accQ[4], accK[4], accV[4];
  const v8f vzero = {0.f, 0.f, 0.f, 0.f, 0.f, 0.f, 0.f, 0.f};
#pragma unroll
  for (int t = 0; t < 4; ++t) { accQ[t] = vzero; accK[t] = vzero; accV[t] = vzero; }

  for (int kc = 0; kc < EMB; kc += KC) {
    __syncthreads();   // previous chunk fully consumed before restaging
    // stage x tile: 16 rows x 32 floats (8 float4/row), per-wave
#pragma unroll
    for (int i = 0; i < 4; ++i) {
      int vi = i * 32 + lane;    // 0..127 float4 slots
      int r  = vi >> 3;
      int c4 = vi & 7;
      const float4 val =
          *(const float4*)(x + (size_t)(row0 + r) * EMB + kc + c4 * 4);
      *(float4*)(&xs[wave][r * XS_STRIDE + c4 * 4]) = val;
    }
    stage_wT(Wq, wt[0], kc, tid);
    stage_wT(Wk, wt[1], kc, tid);
    stage_wT(Wv, wt[2], kc, tid);
    __syncthreads();

    // prefetch next chunk while this one computes (global_prefetch_b8)
    if (kc + KC < EMB) {
      __builtin_prefetch(x + (size_t)(row0 + m) * EMB + kc + KC, 0, 1);
      __builtin_prefetch(Wq + (size_t)(kc + KC + lane) * HD, 0, 1);
      __builtin_prefetch(Wk + (size_t)(kc + KC + lane) * HD, 0, 1);
      __builtin_prefetch(Wv + (size_t)(kc + KC + lane) * HD, 0, 1);
    }

#pragma unroll
    for (int kk = 0; kk < KC; kk += 4) {
      const v2f a = *(const v2f*)&xs[wave][m * XS_STRIDE + kk + 2 * hi];
#pragma unroll
      for (int nt = 0; nt < 4; ++nt) {
        const int cbase = (nt * 16 + m) * WT_STRIDE + kk + 2 * hi;
        const v2f bq = *(const v2f*)&wt[0][cbase];   // single ds_load_b64
        const v2f bk = *(const v2f*)&wt[1][cbase];
        const v2f bv = *(const v2f*)&wt[2][cbase];
        accQ[nt] = wmma4(a, bq, accQ[nt]);
        accK[nt] = wmma4(a, bk, accK[nt]);
        accV[nt] = wmma4(a, bv, accV[nt]);
      }
    }
  }

  // D layout: lane holds rows r+8*hi, col m.  q,k row-major; v transposed.
#pragma unroll
  for (int nt = 0; nt < 4; ++nt)
#pragma unroll
    for (int r = 0; r < 8; ++r) {
      const int srow = row0 + r + 8 * hi;        // 0..16383 = b*2048 + s
      const int col  = nt * 16 + m;
      q[(size_t)srow * HD + col] = accQ[nt][r];
      k[(size_t)srow * HD + col] = accK[nt][r];
      const int bb = srow >> 11;
      const int ss = srow & (SEQ - 1);
      vT[((size_t)bb * HD + col) * SEQ + ss] = accV[nt][r];
    }
}

// ---------------------------------------------------------------------------
// Kernel 2: flash-style attention. One wave per (batch, 16 query rows) tile.
// ---------------------------------------------------------------------------
__global__ __launch_bounds__(128) void attn_kernel(
    const float* __restrict__ q, const float* __restrict__ k,
    const float* __restrict__ vT, float* __restrict__ out) {
  __shared__ float pbuf[4][16 * PB_STRIDE];

  const int tid  = threadIdx.x;
  const int wave = tid >> 5;
  const int lane = tid & 31;
  const int n    = lane & 15;
  const int hi   = lane >> 4;
  const int tile = blockIdx.x * 4 + wave;          // 0..1023
  const int b    = tile >> 7;                      // 128 q-tiles / batch
  const int q0   = (tile & 127) * 16;

  const float* qb  = q  + (size_t)b * SEQ * HD;
  const float* kb  = k  + (size_t)b * SEQ * HD;
  const float* vbT = vT + (size_t)b * HD * SEQ;    // [64][2048]

  // Preload q A-frags for the whole D=64 reduction (16 k-steps of 4)
  v2f aq[16];
#pragma unroll
  for (int t = 0; t < 16; ++t)
    aq[t] = *(const v2f*)(qb + (size_t)(q0 + n) * HD + t * 4 + 2 * hi);

  const v8f vzero = {0.f, 0.f, 0.f, 0.f, 0.f, 0.f, 0.f, 0.f};
  v8f acc[4];
#pragma unroll
  for (int t = 0; t < 4; ++t) acc[t] = vzero;
  float mrow[8], lrow[8];
#pragma unroll
  for (int r = 0; r < 8; ++r) { mrow[r] = -INFINITY; lrow[r] = 0.f; }

  for (int kt = 0; kt < SEQ / 16; ++kt) {
    const int k0 = kt * 16;

    // scores tile: s = q_tile (16x64) * k_tileT (64x16), fp32 WMMA chain
    v8f s = vzero;
#pragma unroll
    for (int t = 0; t < 16; ++t) {
      const v2f bk = *(const v2f*)(kb + (size_t)(k0 + n) * HD + t * 4 + 2 * hi);
      s = wmma4(aq[t], bk, s);
    }

    // online softmax over the 16 columns of each row (16-lane shfl_xor trees)
    float alpha[8];
#pragma unroll
    for (int r = 0; r < 8; ++r) {
      float sv = s[r] * 0.125f;                  // 1/sqrt(64)
      float mx = sv;
      mx = fmaxf(mx, __shfl_xor(mx, 1, 32));
      mx = fmaxf(mx, __shfl_xor(mx, 2, 32));
      mx = fmaxf(mx, __shfl_xor(mx, 4, 32));
      mx = fmaxf(mx, __shfl_xor(mx, 8, 32));
      const float mnew = fmaxf(mrow[r], mx);
      alpha[r] = __expf(mrow[r] - mnew);
      const float pv = __expf(sv - mnew);
      float rs = pv;
      rs += __shfl_xor(rs, 1, 32);
      rs += __shfl_xor(rs, 2, 32);
      rs += __shfl_xor(rs, 4, 32);
      rs += __shfl_xor(rs, 8, 32);
      lrow[r] = lrow[r] * alpha[r] + rs;
      mrow[r] = mnew;
      s[r] = pv;                                 // s now holds p (D layout)
    }
#pragma unroll
    for (int nt = 0; nt < 4; ++nt)
#pragma unroll
      for (int r = 0; r < 8; ++r) acc[nt][r] *= alpha[r];

    // convert p from D-layout to A-layout via LDS bounce
    __syncthreads();
#pragma unroll
    for (int r = 0; r < 8; ++r) pbuf[wave][(r + 8 * hi) * PB_STRIDE + n] = s[r];
    __syncthreads();

    // acc += p (16x16) * v_tile (16x64); v^T makes B frags contiguous float2
#pragma unroll
    for (int t = 0; t < 4; ++t) {
      const v2f ap = *(const v2f*)&pbuf[wave][n * PB_STRIDE + t * 4 + 2 * hi];
#pragma unroll
      for (int nt = 0; nt < 4; ++nt) {
        const v2f bv =
            *(const v2f*)(vbT + (size_t)(nt * 16 + n) * SEQ + k0 + t * 4 + 2 * hi);
        acc[nt] = wmma4(ap, bv, acc[nt]);
      }
    }
  }

  // epilogue: divide by row sums, write out
#pragma unroll
  for (int r = 0; r < 8; ++r) {
    const float inv = 1.0f / lrow[r];
    const size_t row = (size_t)b * SEQ + q0 + r + 8 * hi;
#pragma unroll
    for (int nt = 0; nt < 4; ++nt)
      out[row * HD + nt * 16 + n] = acc[nt][r] * inv;
  }
}

// ---------------------------------------------------------------------------
extern "C" void kernel_launch(void* const* d_in, const int* in_sizes, int n_in,
                              void* d_out, int out_size, void* d_ws,
                              size_t ws_size, hipStream_t stream) {
  const float* x  = (const float*)d_in[0];
  const float* Wq = (const float*)d_in[1];
  const float* Wk = (const float*)d_in[2];
  const float* Wv = (const float*)d_in[3];
  float* out = (float*)d_out;

  float* qws  = (float*)d_ws;                 // [8][2048][64]
  const size_t per = (size_t)NB * SEQ * HD;   // 1,048,576 floats each
  float* kws  = qws + per;                    // [8][2048][64]
  float* vTws = kws + per;                    // [8][64][2048]

  // 16384 rows / 16 per wave / 4 waves per block = 256 blocks
  qkv_proj_kernel<<<256, 128, 0, stream>>>(x, Wq, Wk, Wv, qws, kws, vTws);
  // 8 batches * 128 q-tiles / 4 waves per block = 256 blocks
  attn_kernel<<<256, 128, 0, stream>>>(qws, kws, vTws, out);
}